// AeroForceLoss_38354057953996
// MI455X (gfx1250) — compile-verified
//
#include <hip/hip_runtime.h>
#include <hip/hip_bf16.h>

// ---------------------------------------------------------------------------
// AeroForceLoss for MI455X (gfx1250, wave32).
// Memory-bound streaming reduction (~100 MB traffic, floor ~4.3us @ 23.3TB/s).
// Cross-lane reductions use V_WMMA_F32_16X16X4_F32 with an all-ones B matrix:
// D[m,n] = sum_k A[m,k].  A layout (documented): lane m holds K=0/1, lane m+16
// holds K=2/3; we feed {p, 0} so D row m = p[m] + p[m+16].  C/D layout
// (documented): D VGPR j holds rows j (lanes 0-15) and j+8 (lanes 16-31), so
// summing the 8 D registers + one shfl_xor(16) yields the full 32-lane sum.
// ---------------------------------------------------------------------------

typedef float v2f __attribute__((ext_vector_type(2)));
typedef float v8f __attribute__((ext_vector_type(8)));

__device__ __forceinline__ float wave_sum_wmma(float p) {
    v2f a; a[0] = p;    a[1] = 0.0f;   // only K=0 (lanes 0-15) / K=2 (lanes 16-31) slot used
    v2f b; b[0] = 1.0f; b[1] = 1.0f;   // all-ones B: layout invariant
    v8f c = {};
    v8f d = __builtin_amdgcn_wmma_f32_16x16x4_f32(
        /*neg_a=*/false, a, /*neg_b=*/false, b,
        /*c_mod=*/(short)0, c, /*reuse_a=*/false, /*reuse_b=*/false);
    float t = d[0] + d[1] + d[2] + d[3] + d[4] + d[5] + d[6] + d[7];
    // lanes 0-15: sum of rows 0..7 ; lanes 16-31: sum of rows 8..15
    t += __shfl_xor(t, 16, 32);
    return t;   // total of all 32 lanes, replicated to every lane
}

// ws layout (floats): ws[0] = sum of squared diffs; ws[8 + 4*b + c] = force[b][c]

__global__ void aero_init_kernel(float* __restrict__ ws, const int* __restrict__ pB) {
    const int B = *pB;
    const int n = 8 + 4 * B;
    for (int i = threadIdx.x; i < n; i += blockDim.x) ws[i] = 0.0f;
}

__device__ __forceinline__ void flush_force(float* ws, long long b,
                                            float fx, float fy, float fz) {
    if (b >= 0) {
        atomicAdd(&ws[8 + 4 * b + 0], fx);
        atomicAdd(&ws[8 + 4 * b + 1], fy);
        atomicAdd(&ws[8 + 4 * b + 2], fz);
    }
}

__global__ void aero_partial_kernel(const float4* __restrict__ pred,
                                    const float4* __restrict__ targ,
                                    const float*  __restrict__ nrm,
                                    const float*  __restrict__ area,
                                    const int*    __restrict__ pS,
                                    float*        __restrict__ ws,
                                    long long N, long long chunk) {
    const long long S  = (long long)(*pS);
    const long long lo = (long long)blockIdx.x * chunk;
    if (lo >= N) return;
    long long hi = lo + chunk;
    if (hi > N) hi = N;

    // Does this block's chunk stay inside a single batch segment? (scalar-uniform)
    const bool uniform = (lo / S) == ((hi - 1) / S);

    float sse = 0.0f, fx = 0.0f, fy = 0.0f, fz = 0.0f;

    if (uniform) {
        for (long long i = lo + threadIdx.x; i < hi; i += blockDim.x) {
            float4 p = pred[i];
            float4 t = targ[i];
            float d0 = p.x - t.x, d1 = p.y - t.y, d2 = p.z - t.z, d3 = p.w - t.w;
            sse += d0 * d0 + d1 * d1 + d2 * d2 + d3 * d3;
            float aw = area[i];
            const float* nn = nrm + 3 * i;
            fx += (d0 + d1) * nn[0] * aw;
            fy += (d0 + d2) * nn[1] * aw;
            fz += (d0 + d3) * nn[2] * aw;
        }
    } else {
        // Rare boundary block: flush per-thread force partials per segment.
        long long bcur = -1;
        for (long long i = lo + threadIdx.x; i < hi; i += blockDim.x) {
            long long b = i / S;
            if (b != bcur) {
                flush_force(ws, bcur, fx, fy, fz);
                fx = fy = fz = 0.0f;
                bcur = b;
            }
            float4 p = pred[i];
            float4 t = targ[i];
            float d0 = p.x - t.x, d1 = p.y - t.y, d2 = p.z - t.z, d3 = p.w - t.w;
            sse += d0 * d0 + d1 * d1 + d2 * d2 + d3 * d3;
            float aw = area[i];
            const float* nn = nrm + 3 * i;
            fx += (d0 + d1) * nn[0] * aw;
            fy += (d0 + d2) * nn[1] * aw;
            fz += (d0 + d3) * nn[2] * aw;
        }
        flush_force(ws, bcur, fx, fy, fz);
        fx = fy = fz = 0.0f;   // WMMA stage below then reduces zeros (harmless)
    }

    // ---- per-wave reduction via WMMA (EXEC is all-1s here: no divergence) ----
    float wsse = wave_sum_wmma(sse);
    float wfx  = wave_sum_wmma(fx);
    float wfy  = wave_sum_wmma(fy);
    float wfz  = wave_sum_wmma(fz);

    __shared__ float red[8][4];        // 256 threads = 8 wave32s
    const int wave = threadIdx.x >> 5;
    const int lane = threadIdx.x & 31;
    if (lane == 0) {
        red[wave][0] = wsse;
        red[wave][1] = wfx;
        red[wave][2] = wfy;
        red[wave][3] = wfz;
    }
    __syncthreads();

    if (threadIdx.x == 0) {
        float s0 = 0.0f, s1 = 0.0f, s2 = 0.0f, s3 = 0.0f;
        const int nw = (int)(blockDim.x >> 5);
        for (int w = 0; w < nw; ++w) {
            s0 += red[w][0]; s1 += red[w][1]; s2 += red[w][2]; s3 += red[w][3];
        }
        atomicAdd(&ws[0], s0);
        if (uniform) {
            long long b = lo / S;
            atomicAdd(&ws[8 + 4 * b + 0], s1);
            atomicAdd(&ws[8 + 4 * b + 1], s2);
            atomicAdd(&ws[8 + 4 * b + 2], s3);
        }
    }
}

__global__ void aero_final_kernel(const float* __restrict__ ws,
                                  const int*   __restrict__ pB,
                                  float*       __restrict__ out,
                                  long long N) {
    if (threadIdx.x == 0 && blockIdx.x == 0) {
        const int B = *pB;
        float base = ws[0] / (float)(4LL * N);
        float fsum = 0.0f;
        for (int b = 0; b < B; ++b) {
            float x = ws[8 + 4 * b + 0];
            float y = ws[8 + 4 * b + 1];
            float z = ws[8 + 4 * b + 2];
            fsum += sqrtf(x * x + y * y + z * z);
        }
        out[0] = base + 0.1f * (fsum / (float)B);
    }
}

extern "C" void kernel_launch(void* const* d_in, const int* in_sizes, int n_in,
                              void* d_out, int out_size, void* d_ws, size_t ws_size,
                              hipStream_t stream) {
    const float4* pred = (const float4*)d_in[0];   // [N,4] f32
    const float4* targ = (const float4*)d_in[1];   // [N,4] f32
    const float*  nrm  = (const float*) d_in[2];   // [N,3] f32
    const float*  area = (const float*) d_in[3];   // [N,1] f32
    const int*    pB   = (const int*)   d_in[4];   // scalar batch_size (device)
    const int*    pS   = (const int*)   d_in[5];   // scalar sim_len (device)
    float* ws  = (float*)d_ws;
    float* out = (float*)d_out;

    const long long N = (long long)in_sizes[3];    // areas has N elements

    const int       threads = 256;                 // 8 wave32s / block
    const long long ITERS   = 4;                   // ~4 points per thread
    const long long chunk   = (long long)threads * ITERS;
    long long nblocks = (N + chunk - 1) / chunk;
    if (nblocks < 1) nblocks = 1;

    aero_init_kernel<<<1, 256, 0, stream>>>(ws, pB);
    aero_partial_kernel<<<(int)nblocks, threads, 0, stream>>>(pred, targ, nrm, area,
                                                              pS, ws, N, chunk);
    aero_final_kernel<<<1, 32, 0, stream>>>(ws, pB, out, N);
}